// Normalizer_48636209660399
// MI455X (gfx1250) — compile-verified
//
#include <hip/hip_runtime.h>

#define FRAME_LEN 133
#define SEQ_LEN   2048
#define NO_HAND   (-1.0f)

// --- CDNA5 async global->LDS copy (ASYNCcnt path). The builtin's probed
// signature: (v4i addrspace(1)* gsrc, v4i addrspace(3)* ldst, imm, imm cpol).
// Guarded so absence of the builtin degrades to direct loads. ---
#if defined(__has_builtin)
#  if __has_builtin(__builtin_amdgcn_global_load_async_to_lds_b128) && \
      __has_builtin(__builtin_amdgcn_s_wait_asynccnt)
#    define HAVE_ASYNC_LDS 1
#  endif
#endif
#ifndef HAVE_ASYNC_LDS
#  define HAVE_ASYNC_LDS 0
#endif

#if HAVE_ASYNC_LDS
typedef int v4i __attribute__((ext_vector_type(4)));
typedef __attribute__((address_space(1))) v4i* as1_v4i;
typedef __attribute__((address_space(3))) v4i* as3_v4i;
#endif

// ---------------------------------------------------------------------------
// Phase 1: per batch row, find the LAST frame where each hand is valid and
// emit [ref0x, ref0y, ref0z, any0, ref1x, ref1y, ref1z, any1] to scratch.
// Reads only the 2 flag columns (~4 MB total) -- negligible vs. main stream.
// ---------------------------------------------------------------------------
__global__ __launch_bounds__(256)
void refs_kernel(const float* __restrict__ X, float* __restrict__ refs) {
    __shared__ int sm0[256];
    __shared__ int sm1[256];
    const int b   = blockIdx.x;
    const int tid = threadIdx.x;
    const float* row = X + (size_t)b * SEQ_LEN * FRAME_LEN;

    int best0 = -1, best1 = -1;
    for (int s = tid; s < SEQ_LEN; s += 256) {
        const float* fr = row + (size_t)s * FRAME_LEN;
        if (fr[3]  != NO_HAND) best0 = s;   // s increases -> keeps max
        if (fr[68] != NO_HAND) best1 = s;
    }
    sm0[tid] = best0;
    sm1[tid] = best1;
    __syncthreads();
    for (int off = 128; off > 0; off >>= 1) {
        if (tid < off) {
            sm0[tid] = max(sm0[tid], sm0[tid + off]);
            sm1[tid] = max(sm1[tid], sm1[tid + off]);
        }
        __syncthreads();
    }
    if (tid == 0) {
        int l0 = sm0[0], l1 = sm1[0];
        const float any0 = (l0 >= 0) ? 1.0f : 0.0f;
        const float any1 = (l1 >= 0) ? 1.0f : 0.0f;
        if (l0 < 0) l0 = 0;
        if (l1 < 0) l1 = 0;
        const float* f0 = row + (size_t)l0 * FRAME_LEN;
        const float* f1 = row + (size_t)l1 * FRAME_LEN;
        float* r = refs + (size_t)b * 8;
        r[0] = f0[5];  r[1] = f0[26]; r[2] = f0[47];  r[3] = any0;
        r[4] = f1[70]; r[5] = f1[91]; r[6] = f1[112]; r[7] = any1;
    }
}

// ---------------------------------------------------------------------------
// Phase 2: streaming normalize. One wave32 per frame:
//   lane = column within frame, 5 passes of 32 lanes cover 133 columns.
// Flags (cols 3 and 68) are broadcast from already-loaded registers via
// __shfl (LDS bpermute) -- zero extra memory traffic.
// Refs are staged to LDS once per block via the CDNA5 async global->LDS DMA
// (ASYNCcnt) when available. Non-temporal loads/stores keep the 558 MB
// stream from thrashing the 192 MB L2; global_prefetch_b8 runs ahead.
// ---------------------------------------------------------------------------
__global__ __launch_bounds__(256)
void normalize_kernel(const float* __restrict__ X,
                      const float* __restrict__ refs,
                      float* __restrict__ OUT) {
    const int b    = blockIdx.y;
    const int lane = threadIdx.x & 31;
    const int wave = threadIdx.x >> 5;
    const int FPW  = 16;                       // frames per wave
    const int s0   = blockIdx.x * (8 * FPW) + wave * FPW;

    // Stage the 8 per-batch ref floats into LDS (two 16-byte DMAs, broadcast to block)
    __shared__ __align__(16) float refbuf[8];
#if HAVE_ASYNC_LDS
    if (threadIdx.x == 0) {
        __builtin_amdgcn_global_load_async_to_lds_b128(
            (as1_v4i)(refs + (size_t)b * 8),     (as3_v4i)(refbuf),     0, 0);
        __builtin_amdgcn_global_load_async_to_lds_b128(
            (as1_v4i)(refs + (size_t)b * 8 + 4), (as3_v4i)(refbuf + 4), 0, 0);
        __builtin_amdgcn_s_wait_asynccnt(0);
    }
#else
    if (threadIdx.x < 8)
        refbuf[threadIdx.x] = refs[(size_t)b * 8 + threadIdx.x];
#endif
    __syncthreads();

    const float r0x = refbuf[0], r0y = refbuf[1], r0z = refbuf[2], a0 = refbuf[3];
    const float r1x = refbuf[4], r1y = refbuf[5], r1z = refbuf[6], a1 = refbuf[7];

    // Per-lane column classification for each of the 5 passes (loop-invariant)
    float c0p[5], c1p[5];
#pragma unroll
    for (int i = 0; i < 5; ++i) {
        const int f = lane + 32 * i;
        float c0 = 0.0f, c1 = 0.0f;
        if      (f >= 5  && f <= 25)  c0 = r0x;
        else if (f >= 26 && f <= 46)  c0 = r0y;
        else if (f >= 47 && f <= 67)  c0 = r0z;
        if      (f >= 70 && f <= 90)  c1 = r1x;
        else if (f >= 91 && f <= 111) c1 = r1y;
        else if (f >= 112)            c1 = r1z;  // lanes with f>132 never store
        c0p[i] = c0; c1p[i] = c1;
    }

    const float* row  = X   + (size_t)b * SEQ_LEN * FRAME_LEN;
    float*       orow = OUT + (size_t)b * SEQ_LEN * FRAME_LEN;

    for (int k = 0; k < FPW; ++k) {
        const int s = s0 + k;
        const float* fp = row  + (size_t)s * FRAME_LEN;
        float*       op = orow + (size_t)s * FRAME_LEN;

        // gfx1250 prefetch: cover frame s+8 (5 cachelines) speculatively
        if (s + 8 < SEQ_LEN)
            __builtin_prefetch(fp + 8 * FRAME_LEN + lane * 5, 0, 0);

        // 5 coalesced non-temporal passes over the 133-float frame
        const float v0 = __builtin_nontemporal_load(fp + lane);
        const float v1 = __builtin_nontemporal_load(fp + 32 + lane);
        const float v2 = __builtin_nontemporal_load(fp + 64 + lane);
        const float v3 = __builtin_nontemporal_load(fp + 96 + lane);
        float v4 = 0.0f;
        if (lane < 5)
            v4 = __builtin_nontemporal_load(fp + 128 + lane);

        // Flags live in the data we just loaded: f=3 (pass0 lane3), f=68 (pass2 lane4)
        const float flag0 = __shfl(v0, 3, 32);
        const float flag1 = __shfl(v2, 4, 32);
        const float m0 = (flag0 != NO_HAND) ? a0 : 0.0f;
        const float m1 = (flag1 != NO_HAND) ? a1 : 0.0f;

        __builtin_nontemporal_store(v0 - (m0 * c0p[0] + m1 * c1p[0]), op + lane);
        __builtin_nontemporal_store(v1 - (m0 * c0p[1] + m1 * c1p[1]), op + 32 + lane);
        __builtin_nontemporal_store(v2 - (m0 * c0p[2] + m1 * c1p[2]), op + 64 + lane);
        __builtin_nontemporal_store(v3 - (m0 * c0p[3] + m1 * c1p[3]), op + 96 + lane);
        if (lane < 5)
            __builtin_nontemporal_store(v4 - (m0 * c0p[4] + m1 * c1p[4]), op + 128 + lane);
    }
}

extern "C" void kernel_launch(void* const* d_in, const int* in_sizes, int n_in,
                              void* d_out, int out_size, void* d_ws, size_t ws_size,
                              hipStream_t stream) {
    const float* X    = (const float*)d_in[0];
    float*       OUT  = (float*)d_out;
    float*       refs = (float*)d_ws;           // B * 8 floats of scratch

    const int B = in_sizes[0] / (SEQ_LEN * FRAME_LEN);

    refs_kernel<<<dim3(B), 256, 0, stream>>>(X, refs);

    // 8 waves/block * 16 frames/wave = 128 frames/block; 2048/128 = 16 blocks/row
    dim3 grid(SEQ_LEN / 128, B);
    normalize_kernel<<<grid, 256, 0, stream>>>(X, refs, OUT);
}